// DGC_57105885168305
// MI455X (gfx1250) — compile-verified
//
#include <hip/hip_runtime.h>
#include <stdint.h>

#define BLOCK 256
#define WAVE  32
#define WPB   (BLOCK / WAVE)   // 8 waves per block (wave32)
#define WTILE 512              // floats per wave tile (2KB) staged via TDM

// ---------------------------------------------------------------------------
// CDNA5 Tensor Data Mover path. This toolchain exposes the 6-arg builtin:
//   (uint32x4 g0, int32x8 g1, int32x4 g2, int32x4 g3, int32x8 g4, i32 cpol)
// ---------------------------------------------------------------------------
typedef unsigned int v4u __attribute__((ext_vector_type(4)));
typedef int          v8i __attribute__((ext_vector_type(8)));
typedef int          v4i __attribute__((ext_vector_type(4)));

#if defined(__HIP_DEVICE_COMPILE__) && defined(__gfx1250__) && \
    __has_builtin(__builtin_amdgcn_tensor_load_to_lds) &&      \
    __has_builtin(__builtin_amdgcn_s_wait_tensorcnt)
#define USE_TDM 1
#else
#define USE_TDM 0
#endif

#if USE_TDM
// 1-D tile DMA: global -> LDS, nElems f32 elements. D# per CDNA5 ISA ch.8.
__device__ __forceinline__ void tdm_load_to_lds(unsigned lds_off,
                                                const float* gsrc,
                                                unsigned nElems) {
  unsigned long long ga = (unsigned long long)(uintptr_t)gsrc;
  // Group 0: count=1 | lds_addr | global_addr[56:0] | type=2 (bits 127:126)
  v4u g0 = { 1u,
             lds_off,
             (unsigned)ga,
             (unsigned)(((ga >> 32) & 0x01FFFFFFull) | (2u << 30)) };
  // Group 1: data_size=4B (bits17:16), tensor_dim0=nElems (bits79:48),
  //          tensor_dim1=1 (bits111:80), tile_dim0=nElems (bits127:112),
  //          tile_dim1=1 (bits143:128), tensor_dim0_stride=nElems (bits207:160)
  v8i g1;
  g1[0] = (int)(2u << 16);
  g1[1] = (int)((nElems & 0xFFFFu) << 16);
  g1[2] = (int)(((nElems >> 16) & 0xFFFFu) | (1u << 16));
  g1[3] = (int)((nElems & 0xFFFFu) << 16);
  g1[4] = 1;                 // tile_dim1 = 1
  g1[5] = (int)nElems;       // tensor_dim0_stride
  g1[6] = 0;
  g1[7] = 0;
  v4i z4 = { 0, 0, 0, 0 };
  v8i z8 = { 0, 0, 0, 0, 0, 0, 0, 0 };
  __builtin_amdgcn_tensor_load_to_lds(g0, g1, z4, z4, z8, 0);
}
#endif

__device__ __forceinline__ unsigned magkey(float f) {
  return __float_as_uint(f) & 0x7fffffffu;
}

// ---------------------------------------------------------------------------
// K1 (fused): outRes = r + m*g, outSp = 0, coarse histogram of key>>20.
// Each wave DMA-stages its 2KB grad/residual tiles into LDS via the TDM,
// then consumes them as float4 (ds_load_b128 -> global_store_b128).
// ---------------------------------------------------------------------------
__global__ void dgc_pass1(const float* __restrict__ grad,
                          const float* __restrict__ resid,
                          const float* __restrict__ momp,
                          float* __restrict__ outRes,
                          float* __restrict__ outSp,
                          unsigned* __restrict__ hist1,
                          int N) {
  __shared__ unsigned lhist[2048];
#if USE_TDM
  __shared__ float tg[WPB][WTILE];
  __shared__ float tr[WPB][WTILE];
#endif
  for (int i = threadIdx.x; i < 2048; i += BLOCK) lhist[i] = 0u;
  __syncthreads();

  const float m    = momp[0];
  const int   wid  = threadIdx.x / WAVE;
  const int   lane = threadIdx.x % WAVE;
  const int   nTiles  = (N + WTILE - 1) / WTILE;
  const int   gw      = blockIdx.x * WPB + wid;
  const int   wstride = gridDim.x * WPB;
  const float4 zero4  = make_float4(0.f, 0.f, 0.f, 0.f);

  for (int t = gw; t < nTiles; t += wstride) {
    const int base = t * WTILE;
    const int cnt  = min(WTILE, N - base);
#if USE_TDM
    tdm_load_to_lds((unsigned)(uintptr_t)&tg[wid][0], grad + base,  (unsigned)cnt);
    tdm_load_to_lds((unsigned)(uintptr_t)&tr[wid][0], resid + base, (unsigned)cnt);
    __builtin_amdgcn_s_wait_tensorcnt(0);
    asm volatile("" ::: "memory");
#endif
    if (cnt == WTILE) {
      // full tile: 128-bit path
#if USE_TDM
      const float4* g4 = (const float4*)&tg[wid][0];
      const float4* r4 = (const float4*)&tr[wid][0];
#else
      const float4* g4 = (const float4*)(grad + base);
      const float4* r4 = (const float4*)(resid + base);
#endif
      float4* oR = (float4*)(outRes + base);
      float4* oS = (float4*)(outSp + base);
      for (int q = lane; q < WTILE / 4; q += WAVE) {
        float4 g = g4[q];
        float4 r = r4[q];
        float4 res;
        res.x = r.x + m * g.x;
        res.y = r.y + m * g.y;
        res.z = r.z + m * g.z;
        res.w = r.w + m * g.w;
        oR[q] = res;
        oS[q] = zero4;
        atomicAdd(&lhist[magkey(g.x + r.x) >> 20], 1u);
        atomicAdd(&lhist[magkey(g.y + r.y) >> 20], 1u);
        atomicAdd(&lhist[magkey(g.z + r.z) >> 20], 1u);
        atomicAdd(&lhist[magkey(g.w + r.w) >> 20], 1u);
      }
    } else {
      for (int e = lane; e < cnt; e += WAVE) {
#if USE_TDM
        float g = tg[wid][e];
        float r = tr[wid][e];
#else
        float g = grad[base + e];
        float r = resid[base + e];
#endif
        outRes[base + e] = r + m * g;
        outSp [base + e] = 0.0f;
        atomicAdd(&lhist[magkey(g + r) >> 20], 1u);
      }
    }
  }
  __syncthreads();
  for (int i = threadIdx.x; i < 2048; i += BLOCK)
    if (lhist[i]) atomicAdd(&hist1[i], lhist[i]);
}

// ---------------------------------------------------------------------------
// K0: zero histograms + control block
// ---------------------------------------------------------------------------
__global__ void dgc_zero(unsigned* w, int n) {
  int i = blockIdx.x * blockDim.x + threadIdx.x;
  if (i < n) w[i] = 0u;
}

// ---------------------------------------------------------------------------
// Radix-select scans (tiny single-thread kernels).
// ctrl: [0]=B1 [1]=cnt_gt1 [2]=B2 [3]=cnt_gt2 [4]=T [5]=needTies
//       [6]=selCount [7]=tieCount
// ---------------------------------------------------------------------------
__global__ void dgc_scan1(const unsigned* h, unsigned* ctrl, int k) {
  if (threadIdx.x || blockIdx.x) return;
  unsigned acc = 0;
  for (int b = 2047; b >= 0; --b) {
    unsigned c = h[b];
    if (acc + c >= (unsigned)k) { ctrl[0] = (unsigned)b; ctrl[1] = acc; return; }
    acc += c;
  }
  ctrl[0] = 0u; ctrl[1] = acc;
}

__global__ void dgc_scan2(const unsigned* h, unsigned* ctrl, int k) {
  if (threadIdx.x || blockIdx.x) return;
  unsigned acc = ctrl[1];
  for (int b = 4095; b >= 0; --b) {
    unsigned c = h[b];
    if (acc + c >= (unsigned)k) { ctrl[2] = (unsigned)b; ctrl[3] = acc; return; }
    acc += c;
  }
  ctrl[2] = 0u; ctrl[3] = acc;
}

__global__ void dgc_scan3(const unsigned* h, unsigned* ctrl, int k) {
  if (threadIdx.x || blockIdx.x) return;
  unsigned acc = ctrl[3];
  unsigned B1 = ctrl[0], B2 = ctrl[2];
  for (int b = 255; b >= 0; --b) {
    unsigned c = h[b];
    if (acc + c >= (unsigned)k) {
      ctrl[4] = (B1 << 20) | (B2 << 8) | (unsigned)b;
      ctrl[5] = (unsigned)k - acc;
      return;
    }
    acc += c;
  }
  ctrl[4] = (B1 << 20) | (B2 << 8);
  ctrl[5] = (unsigned)k - acc;
}

// ---------------------------------------------------------------------------
// K3 / K5: refine histograms (full re-stream with b128 loads, recompute flat)
// ---------------------------------------------------------------------------
__global__ void dgc_hist2(const float* __restrict__ g, const float* __restrict__ r,
                          unsigned* __restrict__ hist2, const unsigned* __restrict__ ctrl,
                          int N) {
  unsigned B1 = ctrl[0];
  const float4* g4 = (const float4*)g;
  const float4* r4 = (const float4*)r;
  int n4 = N >> 2;
  int tid = blockIdx.x * blockDim.x + threadIdx.x;
  int stride = gridDim.x * blockDim.x;
  for (int i = tid; i < n4; i += stride) {
    float4 gv = g4[i], rv = r4[i];
    unsigned u0 = magkey(gv.x + rv.x), u1 = magkey(gv.y + rv.y);
    unsigned u2 = magkey(gv.z + rv.z), u3 = magkey(gv.w + rv.w);
    if ((u0 >> 20) == B1) atomicAdd(&hist2[(u0 >> 8) & 0xFFFu], 1u);
    if ((u1 >> 20) == B1) atomicAdd(&hist2[(u1 >> 8) & 0xFFFu], 1u);
    if ((u2 >> 20) == B1) atomicAdd(&hist2[(u2 >> 8) & 0xFFFu], 1u);
    if ((u3 >> 20) == B1) atomicAdd(&hist2[(u3 >> 8) & 0xFFFu], 1u);
  }
  if (tid == 0)
    for (int i = n4 << 2; i < N; ++i) {
      unsigned u = magkey(g[i] + r[i]);
      if ((u >> 20) == B1) atomicAdd(&hist2[(u >> 8) & 0xFFFu], 1u);
    }
}

__global__ void dgc_hist3(const float* __restrict__ g, const float* __restrict__ r,
                          unsigned* __restrict__ hist3, const unsigned* __restrict__ ctrl,
                          int N) {
  unsigned hi = (ctrl[0] << 12) | ctrl[2];
  const float4* g4 = (const float4*)g;
  const float4* r4 = (const float4*)r;
  int n4 = N >> 2;
  int tid = blockIdx.x * blockDim.x + threadIdx.x;
  int stride = gridDim.x * blockDim.x;
  for (int i = tid; i < n4; i += stride) {
    float4 gv = g4[i], rv = r4[i];
    unsigned u0 = magkey(gv.x + rv.x), u1 = magkey(gv.y + rv.y);
    unsigned u2 = magkey(gv.z + rv.z), u3 = magkey(gv.w + rv.w);
    if ((u0 >> 8) == hi) atomicAdd(&hist3[u0 & 0xFFu], 1u);
    if ((u1 >> 8) == hi) atomicAdd(&hist3[u1 & 0xFFu], 1u);
    if ((u2 >> 8) == hi) atomicAdd(&hist3[u2 & 0xFFu], 1u);
    if ((u3 >> 8) == hi) atomicAdd(&hist3[u3 & 0xFFu], 1u);
  }
  if (tid == 0)
    for (int i = n4 << 2; i < N; ++i) {
      unsigned u = magkey(g[i] + r[i]);
      if ((u >> 8) == hi) atomicAdd(&hist3[u & 0xFFu], 1u);
    }
}

// ---------------------------------------------------------------------------
// K7: compact strictly-above-threshold; collect threshold ties separately.
// ---------------------------------------------------------------------------
__device__ __forceinline__ void compact_one(float f, int i, unsigned T,
                                            unsigned* ctrl,
                                            int* selIdx, float* selVal,
                                            int* tieIdx, float* tieVal, int k) {
  unsigned u = magkey(f);
  if (u > T) {
    unsigned s = atomicAdd(&ctrl[6], 1u);
    if (s < (unsigned)k) { selIdx[s] = i; selVal[s] = f; }
  } else if (u == T) {
    unsigned tp = atomicAdd(&ctrl[7], 1u);
    if (tp < 1024u) { tieIdx[tp] = i; tieVal[tp] = f; }
  }
}

__global__ void dgc_compact(const float* __restrict__ g, const float* __restrict__ r,
                            unsigned* __restrict__ ctrl,
                            int* __restrict__ selIdx, float* __restrict__ selVal,
                            int* __restrict__ tieIdx, float* __restrict__ tieVal,
                            int N, int k) {
  unsigned T = ctrl[4];
  const float4* g4 = (const float4*)g;
  const float4* r4 = (const float4*)r;
  int n4 = N >> 2;
  int tid = blockIdx.x * blockDim.x + threadIdx.x;
  int stride = gridDim.x * blockDim.x;
  for (int i = tid; i < n4; i += stride) {
    float4 gv = g4[i], rv = r4[i];
    int b = i << 2;
    compact_one(gv.x + rv.x, b + 0, T, ctrl, selIdx, selVal, tieIdx, tieVal, k);
    compact_one(gv.y + rv.y, b + 1, T, ctrl, selIdx, selVal, tieIdx, tieVal, k);
    compact_one(gv.z + rv.z, b + 2, T, ctrl, selIdx, selVal, tieIdx, tieVal, k);
    compact_one(gv.w + rv.w, b + 3, T, ctrl, selIdx, selVal, tieIdx, tieVal, k);
  }
  if (tid == 0)
    for (int i = n4 << 2; i < N; ++i)
      compact_one(g[i] + r[i], i, T, ctrl, selIdx, selVal, tieIdx, tieVal, k);
}

// Deterministic tie resolution: pick 'need' ties with the smallest indices.
__global__ void dgc_ties(unsigned* ctrl,
                         const int* tieIdx, const float* tieVal,
                         int* selIdx, float* selVal, int k) {
  if (threadIdx.x || blockIdx.x) return;
  int need = (int)ctrl[5];
  int base = (int)ctrl[6];
  int nt   = (int)ctrl[7]; if (nt > 1024) nt = 1024;
  unsigned used[32];
  for (int i = 0; i < 32; ++i) used[i] = 0u;
  int appended = 0;
  for (int s = 0; s < need && s < nt; ++s) {
    int best = -1, bestIdx = 0x7fffffff;
    for (int j = 0; j < nt; ++j) {
      if ((used[j >> 5] >> (j & 31)) & 1u) continue;
      if (tieIdx[j] < bestIdx) { bestIdx = tieIdx[j]; best = j; }
    }
    if (best < 0) break;
    used[best >> 5] |= 1u << (best & 31);
    int slot = base + appended;
    if (slot < k) { selIdx[slot] = tieIdx[best]; selVal[slot] = tieVal[best]; }
    ++appended;
  }
  ctrl[6] = (unsigned)(base + appended);
}

// ---------------------------------------------------------------------------
// Sort the k (value, index) pairs ascending by signed value (bitonic, P=2^m).
// ---------------------------------------------------------------------------
__global__ void dgc_pad(const unsigned* ctrl, const int* selIdx, const float* selVal,
                        float* sv, int* sx, int k, int P) {
  int j = blockIdx.x * blockDim.x + threadIdx.x;
  if (j >= P) return;
  int cnt = (int)ctrl[6]; if (cnt > k) cnt = k;
  if (j < cnt) { sv[j] = selVal[j]; sx[j] = selIdx[j]; }
  else         { sv[j] = __int_as_float(0x7f800000); sx[j] = 0x7fffffff; }
}

__global__ void dgc_bitonic(float* v, int* x, int len, int stride, int P) {
  int t = blockIdx.x * blockDim.x + threadIdx.x;
  if (t >= (P >> 1)) return;
  int i = 2 * t - (t & (stride - 1));
  int j = i + stride;
  bool up = ((i & len) == 0);
  float vi = v[i], vj = v[j];
  if ((vi > vj) == up) {
    v[i] = vj; v[j] = vi;
    int xi = x[i]; x[i] = x[j]; x[j] = xi;
  }
}

__global__ void dgc_write_topk(const float* sv, const int* sx,
                               float* outIdx, float* outVal, int k) {
  int j = blockIdx.x * blockDim.x + threadIdx.x;
  if (j >= k) return;
  outIdx[j] = (float)sx[j];
  outVal[j] = sv[j];
}

__global__ void dgc_scatter(const unsigned* ctrl, const int* selIdx, const float* selVal,
                            float* outRes, float* outSp, int k, int N) {
  int j = blockIdx.x * blockDim.x + threadIdx.x;
  int cnt = (int)ctrl[6]; if (cnt > k) cnt = k;
  if (j >= cnt) return;
  int i = selIdx[j];
  if (i >= 0 && i < N) {
    outRes[i] = 0.0f;       // residual zeroed at top-k (grad also zeroed there)
    outSp[i]  = selVal[j];  // top value = (grad + residual)[i]
  }
}

// ---------------------------------------------------------------------------
extern "C" void kernel_launch(void* const* d_in, const int* in_sizes, int n_in,
                              void* d_out, int out_size, void* d_ws, size_t ws_size,
                              hipStream_t stream) {
  const float* grad  = (const float*)d_in[0];
  const float* resid = (const float*)d_in[1];
  const float* momp  = (const float*)d_in[2];
  (void)d_in[3]; (void)n_in; (void)ws_size;

  const int N = in_sizes[0];
  int k = (int)(((long long)out_size - 2LL * N) / 2);
  if (k < 1) k = 1;

  int P = 1;
  while (P < k) P <<= 1;
  if (P < 2) P = 2;

  float* outIdx = (float*)d_out;
  float* outVal = outIdx + k;
  float* outRes = outIdx + 2 * (size_t)k;
  float* outSp  = outRes + (size_t)N;

  unsigned* w      = (unsigned*)d_ws;
  unsigned* hist1  = w;              // 2048
  unsigned* hist2  = w + 2048;       // 4096
  unsigned* hist3  = w + 6144;       // 256
  unsigned* ctrl   = w + 6400;       // 16
  int*      tieIdx = (int*)(w + 6416);          // 1024
  float*    tieVal = (float*)(w + 7440);        // 1024
  int*      selIdx = (int*)(w + 8464);          // k
  float*    selVal = (float*)(w + 8464 + k);    // k
  float*    sortV  = (float*)(w + 8464 + 2 * (size_t)k);      // P
  int*      sortX  = (int*)(w + 8464 + 2 * (size_t)k + P);    // P

  const int ZN = 8464;  // words to zero (hists + ctrl + tie arrays; sel gated by counters)
  dgc_zero<<<(ZN + 255) / 256, 256, 0, stream>>>(w, ZN);

  dgc_pass1<<<1024, BLOCK, 0, stream>>>(grad, resid, momp, outRes, outSp, hist1, N);
  dgc_scan1<<<1, 1, 0, stream>>>(hist1, ctrl, k);
  dgc_hist2<<<2048, 256, 0, stream>>>(grad, resid, hist2, ctrl, N);
  dgc_scan2<<<1, 1, 0, stream>>>(hist2, ctrl, k);
  dgc_hist3<<<2048, 256, 0, stream>>>(grad, resid, hist3, ctrl, N);
  dgc_scan3<<<1, 1, 0, stream>>>(hist3, ctrl, k);
  dgc_compact<<<2048, 256, 0, stream>>>(grad, resid, ctrl, selIdx, selVal,
                                        tieIdx, tieVal, N, k);
  dgc_ties<<<1, 1, 0, stream>>>(ctrl, tieIdx, tieVal, selIdx, selVal, k);

  dgc_pad<<<(P + 255) / 256, 256, 0, stream>>>(ctrl, selIdx, selVal, sortV, sortX, k, P);
  const int half = P >> 1;
  for (int len = 2; len <= P; len <<= 1)
    for (int stride = len >> 1; stride > 0; stride >>= 1)
      dgc_bitonic<<<(half + 255) / 256, 256, 0, stream>>>(sortV, sortX, len, stride, P);

  dgc_write_topk<<<(k + 255) / 256, 256, 0, stream>>>(sortV, sortX, outIdx, outVal, k);
  dgc_scatter<<<(k + 255) / 256, 256, 0, stream>>>(ctrl, selIdx, selVal,
                                                   outRes, outSp, k, N);
}